// DivFreeField_3882650436970
// MI455X (gfx1250) — compile-verified
//
#include <hip/hip_runtime.h>

typedef float v2f __attribute__((ext_vector_type(2)));
typedef float v8f __attribute__((ext_vector_type(8)));

#define HDIM 256
#define ZSTRIDE 260            // floats per Z row; 260 % 64 == 4 -> conflict-free b64 frags
#define ZSZ (16 * ZSTRIDE)     // one 16-row activation tile
#define WT_STRIDE 260          // Wt[col][k] transposed weight tile row stride
#define WTSZ (16 * WT_STRIDE)

// Per wave: 4 points x 4 streams (primal, 3 jvp tangents) = 16 WMMA rows.
// Per block: 4 waves = 16 points.
extern "C" __global__ void __launch_bounds__(128)
divfree_mlp_jacfwd(const float* __restrict__ X,
                   const float* __restrict__ W1, const float* __restrict__ B1,
                   const float* __restrict__ W2, const float* __restrict__ B2,
                   const float* __restrict__ W3, const float* __restrict__ B3,
                   const float* __restrict__ Wo,
                   float* __restrict__ OUT, int npts)
{
    extern __shared__ float lds[];
    float* Wt = lds;                                   // shared weight N-tile
    const int wave = threadIdx.x >> 5;
    const int lane = threadIdx.x & 31;
    float* Zbuf = lds + WTSZ + wave * (2 * ZSZ);       // per-wave double buffer

    const int row  = lane & 15;        // A-frag row / C-frag column
    const int half = lane >> 4;        // 0: lanes 0-15, 1: lanes 16-31
    const int koff = half * 2;         // fp32 16x16x4 fragment K offset per lane half

    const int basept = blockIdx.x * 16 + wave * 4;

    // ---------------- layer 1 (3 -> 256) primal + tangent seeds -----------------
    {
        float* Z = Zbuf;
        for (int p = 0; p < 4; ++p) {
            int gp = basept + p; if (gp >= npts) gp = npts - 1;
            const float x0 = X[3 * gp + 0], x1 = X[3 * gp + 1], x2 = X[3 * gp + 2];
            for (int h = lane; h < HDIM; h += 32) {
                const float w0 = W1[0 * HDIM + h];
                const float w1 = W1[1 * HDIM + h];
                const float w2 = W1[2 * HDIM + h];
                const float pre = fmaf(x0, w0, fmaf(x1, w1, fmaf(x2, w2, B1[h])));
                const float m = pre > 0.0f ? 1.0f : 0.0f;
                Z[(4 * p + 0) * ZSTRIDE + h] = pre * m;   // h1 = relu(pre1)
                Z[(4 * p + 1) * ZSTRIDE + h] = w0 * m;    // t1_0
                Z[(4 * p + 2) * ZSTRIDE + h] = w1 * m;    // t1_1
                Z[(4 * p + 3) * ZSTRIDE + h] = w2 * m;    // t1_2
            }
        }
    }
    __syncthreads();

    // ---------------- hidden layers: Znext = relu-mask(Zcur @ W + b) ------------
    for (int layer = 0; layer < 2; ++layer) {
        const float* __restrict__ Wm = (layer == 0) ? W2 : W3;
        const float* __restrict__ Bm = (layer == 0) ? B2 : B3;
        float* Zc = Zbuf + (layer & 1) * ZSZ;
        float* Zn = Zbuf + ((layer + 1) & 1) * ZSZ;

        for (int nt = 0; nt < 16; ++nt) {
            const int n0 = nt * 16;
            // cooperative transpose-stage of the weight column block into LDS
            for (int k = (int)(threadIdx.x >> 4); k < HDIM; k += 8) {
                const int c = threadIdx.x & 15;
                Wt[c * WT_STRIDE + k] = Wm[k * HDIM + n0 + c];
            }
            __syncthreads();

            v8f acc = {};
            #pragma unroll
            for (int kt = 0; kt < 64; ++kt) {
                const int k0 = 4 * kt;
                // A 16x4 fp32 frag: lanes 0-15 hold K=k0..k0+1, lanes 16-31 K=k0+2..k0+3
                v2f a = *(const v2f*)(Zc + row * ZSTRIDE + k0 + koff);
                // B 4x16 fp32 frag: same split over K, col = lane & 15
                v2f b = *(const v2f*)(Wt + row * WT_STRIDE + k0 + koff);
                acc = __builtin_amdgcn_wmma_f32_16x16x4_f32(
                          false, a, false, b, (short)0, acc, false, false);
            }

            // epilogue: bias on primal rows, shared relu mask on all 4 streams.
            // acc[v]: row = v (lanes<16) / v+8 (lanes>=16), col = lane&15.
            // rows 4p..4p+3 of one point sit in consecutive acc entries of one lane.
            const float bias = Bm[n0 + row];
            #pragma unroll
            for (int g = 0; g < 2; ++g) {
                const float pre = acc[4 * g] + bias;
                const float m = pre > 0.0f ? 1.0f : 0.0f;
                const int r0 = half * 8 + 4 * g;
                Zn[(r0 + 0) * ZSTRIDE + n0 + row] = pre * m;
                Zn[(r0 + 1) * ZSTRIDE + n0 + row] = acc[4 * g + 1] * m;
                Zn[(r0 + 2) * ZSTRIDE + n0 + row] = acc[4 * g + 2] * m;
                Zn[(r0 + 3) * ZSTRIDE + n0 + row] = acc[4 * g + 3] * m;
            }
            __syncthreads();   // Wt is overwritten next nt iteration
        }
    }

    // ---------------- output: combine Jacobian J[k][m] = t3_k . Wo[:,m] ---------
    // out0 = J[1][0]+J[2][1]; out1 = -J[0][0]+J[2][2]; out2 = -J[0][1]-J[1][2]
    {
        float* Z3 = Zbuf;   // after 2 layers the result is back in buffer 0
        for (int p = 0; p < 4; ++p) {
            float s10 = 0.f, s21 = 0.f, s00 = 0.f, s22 = 0.f, s01 = 0.f, s12 = 0.f;
            for (int h = lane; h < HDIM; h += 32) {
                const float t0 = Z3[(4 * p + 1) * ZSTRIDE + h];
                const float t1 = Z3[(4 * p + 2) * ZSTRIDE + h];
                const float t2 = Z3[(4 * p + 3) * ZSTRIDE + h];
                const float wo0 = Wo[h * 3 + 0];
                const float wo1 = Wo[h * 3 + 1];
                const float wo2 = Wo[h * 3 + 2];
                s10 = fmaf(t1, wo0, s10);   // J[1][0]
                s21 = fmaf(t2, wo1, s21);   // J[2][1]
                s00 = fmaf(t0, wo0, s00);   // J[0][0]
                s22 = fmaf(t2, wo2, s22);   // J[2][2]
                s01 = fmaf(t0, wo1, s01);   // J[0][1]
                s12 = fmaf(t1, wo2, s12);   // J[1][2]
            }
            float o0 = s10 + s21;
            float o1 = s22 - s00;
            float o2 = -s01 - s12;
            #pragma unroll
            for (int d = 16; d >= 1; d >>= 1) {
                o0 += __shfl_xor(o0, d, 32);
                o1 += __shfl_xor(o1, d, 32);
                o2 += __shfl_xor(o2, d, 32);
            }
            const int gp = basept + p;
            if (lane == 0 && gp < npts) {
                OUT[3 * gp + 0] = o0;
                OUT[3 * gp + 1] = o1;
                OUT[3 * gp + 2] = o2;
            }
        }
    }
}

extern "C" void kernel_launch(void* const* d_in, const int* in_sizes, int n_in,
                              void* d_out, int out_size, void* d_ws, size_t ws_size,
                              hipStream_t stream) {
    const float* X  = (const float*)d_in[0];
    const float* W1 = (const float*)d_in[1];
    const float* B1 = (const float*)d_in[2];
    const float* W2 = (const float*)d_in[3];
    const float* B2 = (const float*)d_in[4];
    const float* W3 = (const float*)d_in[5];
    const float* B3 = (const float*)d_in[6];
    const float* Wo = (const float*)d_in[7];
    float* OUT = (float*)d_out;

    const int npts = in_sizes[0] / 3;               // N = 262144
    const int nblocks = (npts + 15) / 16;           // 16 points per 128-thread block
    const size_t shmem = (size_t)(WTSZ + 4 * 2 * ZSZ) * sizeof(float);  // ~146 KB

    hipLaunchKernelGGL(divfree_mlp_jacfwd, dim3(nblocks), dim3(128), shmem, stream,
                       X, W1, B1, W2, B2, W3, B3, Wo, OUT, npts);
}